// GATEncoder_55490977465146
// MI455X (gfx1250) — compile-verified
//
#include <hip/hip_runtime.h>
#include <math.h>

// ---------------------------------------------------------------------------
// GAT 2-layer encoder for MI455X (gfx1250), wave32.
//   N=50000 nodes, E=1.6M edges, IN_C=HID=128, HEADS=4, F_OUT=32.
// GEMM uses exact-fp32 WMMA (v_wmma_f32_16x16x4_f32): full reference precision
// at negligible cost since the edge scatter phase dominates (L2-resident).
// ---------------------------------------------------------------------------

typedef __attribute__((ext_vector_type(2))) float v2f;
typedef __attribute__((ext_vector_type(8))) float v8f;

#define GAT_N      50000
#define GAT_E      1600000
#define GAT_C      128     // in/out channels per layer
#define GAT_HEADS  4
#define GAT_FOUT   32
#define NEG_SLOPE  0.2f

// ---------------------------------------------------------------------------
// h[M,128] = X[M,128] @ W[128,128]   (row-major, fp32, WMMA 16x16x4)
// One wave -> one 16x16 tile; 8 waves/block -> 16 rows x 128 cols.
// ---------------------------------------------------------------------------
__global__ __launch_bounds__(256) void gemm_f32_wmma(
    const float* __restrict__ X, const float* __restrict__ W,
    float* __restrict__ H)
{
    const int wave = threadIdx.x >> 5;        // 0..7 -> column tile
    const int lane = threadIdx.x & 31;
    const int m0   = blockIdx.x * 16;
    const int n0   = wave * 16;

    const int r    = lane & 15;               // row/col within tile
    const int koff = (lane >> 4) * 2;         // K sub-offset per A/B layout

    const float* __restrict__ xrow = X + (size_t)(m0 + r) * GAT_C + koff;
    const float* __restrict__ wcol = W + (size_t)koff * GAT_C + (n0 + r);

    v8f acc = {};
    #pragma unroll
    for (int k = 0; k < GAT_C; k += 4) {
        v2f a; a.x = xrow[k];                       // X[m0+r][k+koff]
               a.y = xrow[k + 1];                   // X[m0+r][k+koff+1]
        v2f b; b.x = wcol[(size_t)k * GAT_C];       // W[k+koff][n0+r]
               b.y = wcol[(size_t)(k + 1) * GAT_C]; // W[k+koff+1][n0+r]
        acc = __builtin_amdgcn_wmma_f32_16x16x4_f32(
            /*neg_a=*/false, a, /*neg_b=*/false, b,
            /*c_mod=*/(short)0, acc, /*reuse_a=*/false, /*reuse_b=*/false);
    }

    // C/D layout: VGPR i -> row m0 + i + (lane>=16 ? 8 : 0), col n0 + (lane&15)
    float* __restrict__ hp = H + (size_t)(m0 + (lane >> 4) * 8) * GAT_C + (n0 + r);
    #pragma unroll
    for (int i = 0; i < 8; ++i)
        hp[(size_t)i * GAT_C] = acc[i];
}

// ---------------------------------------------------------------------------
// a_s[n,h] = dot(h[n,h,:], att_src[h,:]) ; likewise a_d.  One wave per node.
// ---------------------------------------------------------------------------
__global__ __launch_bounds__(256) void att_scores(
    const float* __restrict__ H,
    const float* __restrict__ att_s, const float* __restrict__ att_d,
    float* __restrict__ a_s, float* __restrict__ a_d, int n)
{
    const int node = blockIdx.x * 8 + (threadIdx.x >> 5);
    const int lane = threadIdx.x & 31;
    if (node >= n) return;
    const float* __restrict__ hp = H + (size_t)node * GAT_C;
    #pragma unroll
    for (int h = 0; h < GAT_HEADS; ++h) {
        const float v = hp[h * GAT_FOUT + lane];
        float s = v * att_s[h * GAT_FOUT + lane];
        float d = v * att_d[h * GAT_FOUT + lane];
        #pragma unroll
        for (int off = 16; off > 0; off >>= 1) {
            s += __shfl_down(s, off, 32);
            d += __shfl_down(d, off, 32);
        }
        if (lane == 0) {
            a_s[node * GAT_HEADS + h] = s;
            a_d[node * GAT_HEADS + h] = d;
        }
    }
}

// ---------------------------------------------------------------------------
// init: e_max = -inf, denom = 0, out = broadcast(bias)
// ---------------------------------------------------------------------------
__global__ __launch_bounds__(256) void init_layer(
    float* __restrict__ emax, float* __restrict__ denom,
    float* __restrict__ out, const float* __restrict__ bias, int n)
{
    const int i = blockIdx.x * 256 + threadIdx.x;
    if (i < n * GAT_HEADS) { emax[i] = -INFINITY; denom[i] = 0.0f; }
    if (i < n * GAT_C)     { out[i] = bias[i & (GAT_C - 1)]; }
}

// ---------------------------------------------------------------------------
// Pass A: segment max of leaky_relu(a_s[src]+a_d[dst]) over dst neighborhoods.
// One thread per (edge, head).
// ---------------------------------------------------------------------------
__global__ __launch_bounds__(256) void edge_seg_max(
    const int* __restrict__ src, const int* __restrict__ dst,
    const float* __restrict__ a_s, const float* __restrict__ a_d,
    float* __restrict__ emax, int ne)
{
    const int t = blockIdx.x * 256 + threadIdx.x;
    if (t >= ne * GAT_HEADS) return;
    const int e = t >> 2, h = t & 3;
    const int s = src[e], d = dst[e];
    float v = a_s[s * GAT_HEADS + h] + a_d[d * GAT_HEADS + h];
    v = (v > 0.0f) ? v : NEG_SLOPE * v;
    atomicMax(&emax[d * GAT_HEADS + h], v);   // global_atomic_max_num_f32
}

// ---------------------------------------------------------------------------
// Pass B: denom[d,h] += exp(e - e_max[d,h])   (logits recomputed; L2 hits)
// ---------------------------------------------------------------------------
__global__ __launch_bounds__(256) void edge_seg_sumexp(
    const int* __restrict__ src, const int* __restrict__ dst,
    const float* __restrict__ a_s, const float* __restrict__ a_d,
    const float* __restrict__ emax, float* __restrict__ denom, int ne)
{
    const int t = blockIdx.x * 256 + threadIdx.x;
    if (t >= ne * GAT_HEADS) return;
    const int e = t >> 2, h = t & 3;
    const int s = src[e], d = dst[e];
    float v = a_s[s * GAT_HEADS + h] + a_d[d * GAT_HEADS + h];
    v = (v > 0.0f) ? v : NEG_SLOPE * v;
    atomicAdd(&denom[d * GAT_HEADS + h], __expf(v - emax[d * GAT_HEADS + h]));
}

// ---------------------------------------------------------------------------
// Pass C: out[dst] += alpha * h[src].  One wave per edge; lane = feature,
// 4-head loop. Scalars (a_s/a_d/emax/denom) are wave-uniform broadcast loads.
// ---------------------------------------------------------------------------
__global__ __launch_bounds__(256) void edge_aggregate(
    const int* __restrict__ src, const int* __restrict__ dst,
    const float* __restrict__ a_s, const float* __restrict__ a_d,
    const float* __restrict__ emax, const float* __restrict__ denom,
    const float* __restrict__ H, float* __restrict__ out, int ne)
{
    const int e = blockIdx.x * 8 + (threadIdx.x >> 5);
    if (e >= ne) return;
    const int lane = threadIdx.x & 31;
    const int s = src[e], d = dst[e];
    #pragma unroll
    for (int h = 0; h < GAT_HEADS; ++h) {
        float v = a_s[s * GAT_HEADS + h] + a_d[d * GAT_HEADS + h];
        v = (v > 0.0f) ? v : NEG_SLOPE * v;
        float den = denom[d * GAT_HEADS + h];
        den = (den > 0.0f) ? den : 1.0f;
        const float alpha = __expf(v - emax[d * GAT_HEADS + h]) / den;
        const float hv = H[(size_t)s * GAT_C + h * GAT_FOUT + lane];
        atomicAdd(&out[(size_t)d * GAT_C + h * GAT_FOUT + lane], alpha * hv);
    }
}

// ---------------------------------------------------------------------------
// In-place ReLU between layers.
// ---------------------------------------------------------------------------
__global__ __launch_bounds__(256) void relu_inplace(float* __restrict__ x, int n)
{
    const int i = blockIdx.x * 256 + threadIdx.x;
    if (i < n) x[i] = fmaxf(x[i], 0.0f);
}

// ---------------------------------------------------------------------------
extern "C" void kernel_launch(void* const* d_in, const int* in_sizes, int n_in,
                              void* d_out, int out_size, void* d_ws, size_t ws_size,
                              hipStream_t stream)
{
    const float* x    = (const float*)d_in[0];
    const int*   eidx = (const int*)  d_in[1];
    const float* W0   = (const float*)d_in[2];
    const float* as0  = (const float*)d_in[3];
    const float* ad0  = (const float*)d_in[4];
    const float* b0   = (const float*)d_in[5];
    const float* W1   = (const float*)d_in[6];
    const float* as1  = (const float*)d_in[7];
    const float* ad1  = (const float*)d_in[8];
    const float* b1   = (const float*)d_in[9];

    const int N = in_sizes[0] / GAT_C;     // 50000
    const int E = in_sizes[1] / 2;         // 1600000
    const int* src = eidx;
    const int* dst = eidx + E;

    // Workspace layout
    char* ws = (char*)d_ws;
    float* hbuf  = (float*)ws;                                  ws += (size_t)N * GAT_C * 4;
    float* out0  = (float*)ws;                                  ws += (size_t)N * GAT_C * 4;
    float* a_s   = (float*)ws;                                  ws += (size_t)N * GAT_HEADS * 4;
    float* a_d   = (float*)ws;                                  ws += (size_t)N * GAT_HEADS * 4;
    float* emax  = (float*)ws;                                  ws += (size_t)N * GAT_HEADS * 4;
    float* denom = (float*)ws;

    const int blk = 256;
    const dim3 gGemm(N / 16);                       // 3125
    const dim3 gNode((N + 7) / 8);                  // wave per node
    const dim3 gInit((N * GAT_C + blk - 1) / blk);  // covers N*128 (and N*4)
    const dim3 gEdgeH((E * GAT_HEADS + blk - 1) / blk);
    const dim3 gEdgeW((E + 7) / 8);                 // wave per edge
    const dim3 gElem((N * GAT_C + blk - 1) / blk);

    // ---------------- Layer 0 ----------------
    init_layer    <<<gInit,  blk, 0, stream>>>(emax, denom, out0, b0, N);
    gemm_f32_wmma <<<gGemm,  blk, 0, stream>>>(x, W0, hbuf);
    att_scores    <<<gNode,  blk, 0, stream>>>(hbuf, as0, ad0, a_s, a_d, N);
    edge_seg_max  <<<gEdgeH, blk, 0, stream>>>(src, dst, a_s, a_d, emax, E);
    edge_seg_sumexp<<<gEdgeH,blk, 0, stream>>>(src, dst, a_s, a_d, emax, denom, E);
    edge_aggregate<<<gEdgeW, blk, 0, stream>>>(src, dst, a_s, a_d, emax, denom,
                                               hbuf, out0, E);
    relu_inplace  <<<gElem,  blk, 0, stream>>>(out0, N * GAT_C);

    // ---------------- Layer 1 ----------------
    float* outf = (float*)d_out;
    init_layer    <<<gInit,  blk, 0, stream>>>(emax, denom, outf, b1, N);
    gemm_f32_wmma <<<gGemm,  blk, 0, stream>>>(out0, W1, hbuf);
    att_scores    <<<gNode,  blk, 0, stream>>>(hbuf, as1, ad1, a_s, a_d, N);
    edge_seg_max  <<<gEdgeH, blk, 0, stream>>>(src, dst, a_s, a_d, emax, E);
    edge_seg_sumexp<<<gEdgeH,blk, 0, stream>>>(src, dst, a_s, a_d, emax, denom, E);
    edge_aggregate<<<gEdgeW, blk, 0, stream>>>(src, dst, a_s, a_d, emax, denom,
                                               hbuf, outf, E);
}